// MTGNNLayer_15066745275106
// MI455X (gfx1250) — compile-verified
//
#include <hip/hip_runtime.h>
#include <hip/hip_bf16.h>

// ---------------- problem constants (from reference) ----------------
#define B_     64
#define C_     32      // C_RES == C_CONV
#define N_     207
#define T_     128
#define TO_    122     // T_OUT
#define CSKIP_ 64
#define BETA_  0.05f

typedef __attribute__((ext_vector_type(2))) float v2f;
typedef __attribute__((ext_vector_type(8))) float v8f;

static __device__ __forceinline__ v8f wmma_f32_4(v2f a, v2f b, v8f c) {
  // D = A(16x4) * B(4x16) + C, fp32, wave32.  emits v_wmma_f32_16x16x4_f32
  return __builtin_amdgcn_wmma_f32_16x16x4_f32(false, a, false, b, (short)0, c,
                                               false, false);
}

// ---------------- workspace layout (floats) ----------------
// Adjacency padded to 256 rows x 208 cols (zeros outside 207x207) so the
// 4-tile M-blocked hop kernel can read A rows up to 255 safely.
constexpr size_t ASZ     = 256 * 208;
constexpr size_t OFF_A1  = 0;
constexpr size_t OFF_A2  = OFF_A1 + ASZ;
constexpr size_t OFF_WI  = OFF_A2 + ASZ;            // 64x224 packed inception weights
constexpr size_t OFF_BI  = OFF_WI + 64 * 224;       // 64 inception biases
constexpr size_t OFF_WS  = OFF_BI + 64;             // 64x4096 padded skip weights
constexpr size_t OFF_WC  = OFF_WS + 64 * 4096;      // 32x32 combined part-0 mlp weight
constexpr size_t OFF_W11 = OFF_WC + 1024;
constexpr size_t OFF_W12 = OFF_W11 + 1024;
constexpr size_t OFF_W21 = OFF_W12 + 1024;
constexpr size_t OFF_W22 = OFF_W21 + 1024;
constexpr size_t OFF_BC  = OFF_W22 + 1024;          // 32 combined bias
constexpr size_t OFF_ST  = OFF_BC + 32;             // 64x2 layernorm stats
constexpr size_t OFF_H   = OFF_ST + 128;
constexpr size_t HELEMS  = (size_t)B_ * C_ * N_ * TO_;   // 51,720,192
constexpr size_t HBUF    = HELEMS + 4096;                // slack: hop K-pad reads row 207
constexpr size_t OFF_P1  = OFF_H + HBUF;
constexpr size_t OFF_P2  = OFF_P1 + HBUF;

constexpr size_t MAIN_OUT = HELEMS;                       // (B,32,N,122)
constexpr int    NT       = N_ * TO_;                     // 25254

// ==================== prep kernels ====================

// Row-normalized A_hat = A (+I), padded to 256x208 with zeros.
__global__ void prep_adj_k(const float* __restrict__ A, float* __restrict__ A1,
                           float* __restrict__ A2) {
  const int v = threadIdx.x;  // 0..255
  float* dst = (blockIdx.x == 0) ? A1 : A2;
  const bool tr = (blockIdx.x == 1);
  if (v >= N_) {
    for (int w = 0; w < 208; ++w) dst[v * 208 + w] = 0.f;
    return;
  }
  float s = 0.f;
  for (int w = 0; w < N_; ++w) {
    float a = tr ? A[w * N_ + v] : A[v * N_ + w];
    if (w == v) a += 1.f;
    s += a;
  }
  const float inv = 1.f / s;
  for (int w = 0; w < 208; ++w) {
    float a = 0.f;
    if (w < N_) {
      a = tr ? A[w * N_ + v] : A[v * N_ + w];
      if (w == v) a += 1.f;
    }
    dst[v * 208 + w] = a * inv;
  }
}

// Pack the 4 inception branches (filter rows 0..31, gate rows 32..63) into a
// dense 64x224 matrix, K-order kk = d*32 + cin (zero-padded taps d < 7-k).
__global__ void prep_incep_k(const float* fw0, const float* fw1, const float* fw2,
                             const float* fw3, const float* fb0, const float* fb1,
                             const float* fb2, const float* fb3, const float* gw0,
                             const float* gw1, const float* gw2, const float* gw3,
                             const float* gb0, const float* gb1, const float* gb2,
                             const float* gb3, float* __restrict__ Wout,
                             float* __restrict__ bout) {
  const int m = threadIdx.x;
  if (m >= 64) return;
  const int kern[4] = {2, 3, 6, 7};
  const bool gate = m >= 32;
  const int ch = m & 31, br = ch >> 3, cc = ch & 7, kk = kern[br];
  const float* w = gate ? (br == 0 ? gw0 : br == 1 ? gw1 : br == 2 ? gw2 : gw3)
                        : (br == 0 ? fw0 : br == 1 ? fw1 : br == 2 ? fw2 : fw3);
  const float* bb = gate ? (br == 0 ? gb0 : br == 1 ? gb1 : br == 2 ? gb2 : gb3)
                         : (br == 0 ? fb0 : br == 1 ? fb1 : br == 2 ? fb2 : fb3);
  bout[m] = bb[cc];
  for (int d = 0; d < 7; ++d)
    for (int cin = 0; cin < 32; ++cin) {
      const int j = d - (7 - kk);
      Wout[(size_t)m * 224 + d * 32 + cin] =
          (j >= 0) ? w[(cc * 32 + cin) * kk + j] : 0.f;
    }
}

// Pad skip weights (64 x 32*122) to K = 32*128 so K decomposes as c=kk>>7, t=kk&127.
__global__ void prep_skipw_k(const float* __restrict__ skip_w, float* __restrict__ Ws) {
  const int cs = blockIdx.x;
  for (int kk = threadIdx.x; kk < 4096; kk += blockDim.x) {
    const int c = kk >> 7, tt = kk & 127;
    Ws[(size_t)cs * 4096 + kk] =
        (tt < TO_) ? skip_w[((size_t)cs * C_ + c) * TO_ + tt] : 0.f;
  }
}

// Slice the two (32x96) 1x1-conv weights into five 32x32 blocks; fold the
// shared part-0 contribution (both mixprops see the same h) into one matrix.
__global__ void prep_mlpw_k(const float* mp1w, const float* mp1b, const float* mp2w,
                            const float* mp2b, float* Wc, float* W11, float* W12,
                            float* W21, float* W22, float* bc) {
  const int co = threadIdx.x;
  if (co >= 32) return;
  bc[co] = mp1b[co] + mp2b[co];
  for (int ci = 0; ci < 32; ++ci) {
    Wc [co * 32 + ci] = mp1w[co * 96 + ci] + mp2w[co * 96 + ci];
    W11[co * 32 + ci] = mp1w[co * 96 + 32 + ci];
    W12[co * 32 + ci] = mp1w[co * 96 + 64 + ci];
    W21[co * 32 + ci] = mp2w[co * 96 + 32 + ci];
    W22[co * 32 + ci] = mp2w[co * 96 + 64 + ci];
  }
}

// ==================== WMMA compute kernels (1 wave / block) ====================

// Inception + gating: 64x224 @ 224x16 implicit GEMM (4 m-tiles per B load),
// fused tanh*sigmoid epilogue.  grid = (8 t-tiles, 207 n, 64 b), block = 32.
__global__ void incep_wmma_k(const float* __restrict__ X, const float* __restrict__ Wf,
                             const float* __restrict__ bias64, float* __restrict__ h) {
  const int t0 = blockIdx.x * 16, n = blockIdx.y, b = blockIdx.z;
  const int l = threadIdx.x, col = l & 15, half = l >> 4;
  const int t = t0 + col;
  const int tl = (t < TO_) ? t : (TO_ - 1);  // clamp loads, mask stores
  const float* xbn = X + (size_t)b * C_ * N_ * T_ + (size_t)n * T_;
  v8f acc0 = {}, acc1 = {}, acc2 = {}, acc3 = {};
  for (int kt = 0; kt < 56; ++kt) {
    const int d = kt >> 3;
    const int cin = (kt & 7) * 4 + half * 2;  // B-tile row for this lane
    v2f bv;
    bv.x = xbn[(size_t)cin * (N_ * T_) + tl + d];
    bv.y = xbn[(size_t)(cin + 1) * (N_ * T_) + tl + d];
    const float* ab = Wf + kt * 4 + half * 2;
    v2f a0, a1, a2, a3;
    a0.x = ab[(size_t)(col) * 224];      a0.y = ab[(size_t)(col) * 224 + 1];
    a1.x = ab[(size_t)(16 + col) * 224]; a1.y = ab[(size_t)(16 + col) * 224 + 1];
    a2.x = ab[(size_t)(32 + col) * 224]; a2.y = ab[(size_t)(32 + col) * 224 + 1];
    a3.x = ab[(size_t)(48 + col) * 224]; a3.y = ab[(size_t)(48 + col) * 224 + 1];
    acc0 = wmma_f32_4(a0, bv, acc0);  // filter ch 0..15
    acc1 = wmma_f32_4(a1, bv, acc1);  // filter ch 16..31
    acc2 = wmma_f32_4(a2, bv, acc2);  // gate   ch 0..15
    acc3 = wmma_f32_4(a3, bv, acc3);  // gate   ch 16..31
  }
  if (t < TO_) {
    for (int r = 0; r < 8; ++r) {
      const int m = half * 8 + r;
      const float f0 = acc0[r] + bias64[m];
      const float f1 = acc1[r] + bias64[16 + m];
      const float g0 = acc2[r] + bias64[32 + m];
      const float g1 = acc3[r] + bias64[48 + m];
      h[(((size_t)b * C_ + m) * N_ + n) * TO_ + t] =
          tanhf(f0) * (1.f / (1.f + expf(-g0)));
      h[(((size_t)b * C_ + 16 + m) * N_ + n) * TO_ + t] =
          tanhf(f1) * (1.f / (1.f + expf(-g1)));
    }
  }
}

// One mixprop hop: hout = beta*hbase + (1-beta) * (A @ hin) along the node dim.
// M-blocked: each wave computes up to 4 v-tiles (64 rows) per B load, since the
// B stream (activations) is the HBM-critical operand and A is L2-resident.
// Tile count is wave-uniform so EXEC stays all-1s at every WMMA.
// grid = (8 t-tiles, 4 v-blocks, B*C planes), block = 32.
__global__ void hop_wmma_k(const float* __restrict__ Apad, const float* __restrict__ hin,
                           const float* __restrict__ hbase, float* __restrict__ hout) {
  const int t0 = blockIdx.x * 16, vb = blockIdx.y * 64;
  const size_t plane = (size_t)blockIdx.z * (N_ * TO_);
  const int l = threadIdx.x, col = l & 15, half = l >> 4;
  const int t = t0 + col;
  const int tl = (t < TO_) ? t : (TO_ - 1);
  const float* hinp = hin + plane;
  int nt4 = (N_ - vb + 15) >> 4;          // remaining 16-row tiles (uniform)
  const int ntiles = nt4 > 4 ? 4 : nt4;   // vb=0/64/128 -> 4, vb=192 -> 1
  v8f acc[4] = {};
  for (int kt = 0; kt < 52; ++kt) {
    const int w = kt * 4 + half * 2;      // K-pad cols >=207 are zero in Apad
    v2f bv;
    bv.x = hinp[(size_t)w * TO_ + tl];
    bv.y = hinp[(size_t)(w + 1) * TO_ + tl];
    if (kt + 2 < 52)  // gfx1250 global_prefetch on the strided B stream
      __builtin_prefetch(&hinp[(size_t)(w + 8) * TO_ + tl], 0, 1);
#pragma unroll
    for (int mt = 0; mt < 4; ++mt) {
      if (mt < ntiles) {
        v2f av;
        const float* ar =
            Apad + (size_t)(vb + mt * 16 + col) * 208 + kt * 4 + half * 2;
        av.x = ar[0];
        av.y = ar[1];
        acc[mt] = wmma_f32_4(av, bv, acc[mt]);
      }
    }
  }
  if (t < TO_) {
    for (int mt = 0; mt < ntiles; ++mt)
      for (int r = 0; r < 8; ++r) {
        const int v = vb + mt * 16 + half * 8 + r;
        if (v < N_) {
          const size_t idx = plane + (size_t)v * TO_ + t;
          hout[idx] = BETA_ * hbase[idx] + (1.f - BETA_) * acc[mt][r];
        }
      }
  }
}

// 1x1 conv slice: out(32 x cols) (+)= W(32x32) @ hin(32 x cols) per batch elem.
// Both 16-row m-tiles computed per B load.  initMode: write with combined
// bias + residual X[..., 6+t]; else accumulate.
// grid = (1579 col-tiles, 1, 64 b), block = 32.
__global__ void mlp_wmma_k(const float* __restrict__ W32, const float* __restrict__ hin,
                           float* __restrict__ out, const float* __restrict__ bias,
                           const float* __restrict__ Xres, int initMode) {
  const int col0 = blockIdx.x * 16, b = blockIdx.z;
  const int l = threadIdx.x, cw = l & 15, half = l >> 4;
  const int col = col0 + cw;
  const int cl = (col < NT) ? col : (NT - 1);
  const float* hb = hin + (size_t)b * C_ * NT;
  v8f acc0 = {}, acc1 = {};
  for (int kt = 0; kt < 8; ++kt) {
    const int k = kt * 4 + half * 2;
    v2f bv, a0, a1;
    bv.x = hb[(size_t)k * NT + cl];
    bv.y = hb[(size_t)(k + 1) * NT + cl];
    a0.x = W32[cw * 32 + k];
    a0.y = W32[cw * 32 + k + 1];
    a1.x = W32[(16 + cw) * 32 + k];
    a1.y = W32[(16 + cw) * 32 + k + 1];
    acc0 = wmma_f32_4(a0, bv, acc0);
    acc1 = wmma_f32_4(a1, bv, acc1);
  }
  if (col < NT) {
    const int n = col / TO_, tt = col % TO_;
    for (int r = 0; r < 8; ++r) {
      const int co0 = half * 8 + r, co1 = 16 + co0;
      const size_t o0 = ((size_t)b * C_ + co0) * NT + col;
      const size_t o1 = ((size_t)b * C_ + co1) * NT + col;
      float v0 = acc0[r], v1 = acc1[r];
      if (initMode) {
        v0 += bias[co0] +
              Xres[(((size_t)b * C_ + co0) * N_ + n) * T_ + (T_ - TO_) + tt];
        v1 += bias[co1] +
              Xres[(((size_t)b * C_ + co1) * N_ + n) * T_ + (T_ - TO_) + tt];
        out[o0] = v0;
        out[o1] = v1;
      } else {
        out[o0] += v0;
        out[o1] += v1;
      }
    }
  }
}

// Skip conv: (64 x 4096-padded) @ (4096 x (b,n)) GEMM, all 4 m-tiles per B
// load; epilogue adds bias + X_skip.  grid = (828 col-tiles), block = 32.
__global__ void skip_wmma_k(const float* __restrict__ Ws, const float* __restrict__ h,
                            const float* __restrict__ skip_b,
                            const float* __restrict__ Xskip, float* __restrict__ oskip) {
  const int col0 = blockIdx.x * 16;
  const int l = threadIdx.x, cw = l & 15, half = l >> 4;
  const int col = col0 + cw;  // flat (b,n); 13248 = 828*16 exactly
  const int b = col / N_, n = col % N_;
  const float* hb = h + ((size_t)b * C_ * N_ + n) * TO_;
  v8f acc[4] = {};
  for (int kt = 0; kt < 1024; ++kt) {
    const int kk = kt * 4 + half * 2;
    const int c0 = kk >> 7, t0 = kk & 127;
    const int c1 = (kk + 1) >> 7, t1 = (kk + 1) & 127;
    v2f bv;
    bv.x = (t0 < TO_) ? hb[(size_t)c0 * (N_ * TO_) + t0] : 0.f;
    bv.y = (t1 < TO_) ? hb[(size_t)c1 * (N_ * TO_) + t1] : 0.f;
#pragma unroll
    for (int mt = 0; mt < 4; ++mt) {
      v2f av;
      const float* ar = Ws + (size_t)(mt * 16 + cw) * 4096 + kk;
      av.x = ar[0];
      av.y = ar[1];
      acc[mt] = wmma_f32_4(av, bv, acc[mt]);
    }
  }
  for (int mt = 0; mt < 4; ++mt)
    for (int r = 0; r < 8; ++r) {
      const int cs = mt * 16 + half * 8 + r;
      const size_t oidx = ((size_t)b * CSKIP_ + cs) * N_ + n;
      oskip[oidx] = acc[mt][r] + skip_b[cs] + Xskip[oidx];
    }
}

// ==================== layernorm ====================

__global__ void stats_k(const float* __restrict__ out, float* __restrict__ stats) {
  const int b = blockIdx.x;
  const size_t base = (size_t)b * (C_ * NT);
  const size_t total = (size_t)C_ * NT;  // 808128
  float s = 0.f, s2 = 0.f;
  for (size_t i = threadIdx.x; i < total; i += blockDim.x) {
    const float v = out[base + i];
    s += v;
    s2 += v * v;
  }
  __shared__ float sh[256], sh2[256];
  sh[threadIdx.x] = s;
  sh2[threadIdx.x] = s2;
  __syncthreads();
  for (int st = blockDim.x / 2; st > 0; st >>= 1) {
    if ((int)threadIdx.x < st) {
      sh[threadIdx.x] += sh[threadIdx.x + st];
      sh2[threadIdx.x] += sh2[threadIdx.x + st];
    }
    __syncthreads();
  }
  if (threadIdx.x == 0) {
    const float mean = sh[0] / (float)total;
    const float var = sh2[0] / (float)total - mean * mean;
    stats[b * 2] = mean;
    stats[b * 2 + 1] = rsqrtf(var + 1e-5f);
  }
}

__global__ void norm_k(float* __restrict__ out, const float* __restrict__ stats,
                       const float* __restrict__ lnw, const float* __restrict__ lnb,
                       const int* __restrict__ idx) {
  const size_t i = (size_t)blockIdx.x * blockDim.x + threadIdx.x;
  const size_t per = (size_t)C_ * NT;
  if (i >= (size_t)B_ * per) return;
  const int b = (int)(i / per);
  const size_t r = i % per;
  const int c = (int)(r / NT);
  const int nt = (int)(r % NT);
  const int n = nt / TO_, t = nt % TO_;
  const int gn = idx[n];
  const size_t widx = ((size_t)c * N_ + gn) * TO_ + t;
  const float v = (out[i] - stats[b * 2]) * stats[b * 2 + 1];
  out[i] = v * lnw[widx] + lnb[widx];
}

// ==================== driver ====================

extern "C" void kernel_launch(void* const* d_in, const int* in_sizes, int n_in,
                              void* d_out, int out_size, void* d_ws, size_t ws_size,
                              hipStream_t stream) {
  const float* X      = (const float*)d_in[0];
  const float* Xskip  = (const float*)d_in[1];
  const float* A      = (const float*)d_in[2];
  const int*   idx    = (const int*)d_in[3];
  const float* fw[4]  = {(const float*)d_in[4], (const float*)d_in[5],
                         (const float*)d_in[6], (const float*)d_in[7]};
  const float* fb[4]  = {(const float*)d_in[8], (const float*)d_in[9],
                         (const float*)d_in[10], (const float*)d_in[11]};
  const float* gw[4]  = {(const float*)d_in[12], (const float*)d_in[13],
                         (const float*)d_in[14], (const float*)d_in[15]};
  const float* gb[4]  = {(const float*)d_in[16], (const float*)d_in[17],
                         (const float*)d_in[18], (const float*)d_in[19]};
  const float* skip_w = (const float*)d_in[20];
  const float* skip_b = (const float*)d_in[21];
  const float* mp1w   = (const float*)d_in[22];
  const float* mp1b   = (const float*)d_in[23];
  const float* mp2w   = (const float*)d_in[24];
  const float* mp2b   = (const float*)d_in[25];
  const float* lnw    = (const float*)d_in[26];
  const float* lnb    = (const float*)d_in[27];

  float* ws   = (float*)d_ws;
  float* A1   = ws + OFF_A1;
  float* A2   = ws + OFF_A2;
  float* Wi   = ws + OFF_WI;
  float* bi   = ws + OFF_BI;
  float* Wsk  = ws + OFF_WS;
  float* Wc   = ws + OFF_WC;
  float* W11  = ws + OFF_W11;
  float* W12  = ws + OFF_W12;
  float* W21  = ws + OFF_W21;
  float* W22  = ws + OFF_W22;
  float* bc   = ws + OFF_BC;
  float* st   = ws + OFF_ST;
  float* h    = ws + OFF_H;
  float* p1   = ws + OFF_P1;
  float* p2   = ws + OFF_P2;

  float* out     = (float*)d_out;
  float* outSkip = out + MAIN_OUT;

  // --- prep ---
  prep_adj_k<<<2, 256, 0, stream>>>(A, A1, A2);
  prep_incep_k<<<1, 64, 0, stream>>>(fw[0], fw[1], fw[2], fw[3], fb[0], fb[1],
                                     fb[2], fb[3], gw[0], gw[1], gw[2], gw[3],
                                     gb[0], gb[1], gb[2], gb[3], Wi, bi);
  prep_skipw_k<<<64, 128, 0, stream>>>(skip_w, Wsk);
  prep_mlpw_k<<<1, 32, 0, stream>>>(mp1w, mp1b, mp2w, mp2b, Wc, W11, W12, W21,
                                    W22, bc);

  // --- inception + gating -> h ---
  incep_wmma_k<<<dim3(8, N_, B_), 32, 0, stream>>>(X, Wi, bi, h);

  // --- skip connection output ---
  skip_wmma_k<<<dim3(828), 32, 0, stream>>>(Wsk, h, skip_b, Xskip, outSkip);

  // --- combined part-0 of both mixprops + bias + residual (writes out) ---
  mlp_wmma_k<<<dim3(1579, 1, B_), 32, 0, stream>>>(Wc, h, out, bc, X, 1);

  // --- mixprop 1 (A) ---
  hop_wmma_k<<<dim3(8, 4, B_ * C_), 32, 0, stream>>>(A1, h, h, p1);
  hop_wmma_k<<<dim3(8, 4, B_ * C_), 32, 0, stream>>>(A1, p1, h, p2);
  mlp_wmma_k<<<dim3(1579, 1, B_), 32, 0, stream>>>(W11, p1, out, nullptr, nullptr, 0);
  mlp_wmma_k<<<dim3(1579, 1, B_), 32, 0, stream>>>(W12, p2, out, nullptr, nullptr, 0);

  // --- mixprop 2 (A^T) ---
  hop_wmma_k<<<dim3(8, 4, B_ * C_), 32, 0, stream>>>(A2, h, h, p1);
  hop_wmma_k<<<dim3(8, 4, B_ * C_), 32, 0, stream>>>(A2, p1, h, p2);
  mlp_wmma_k<<<dim3(1579, 1, B_), 32, 0, stream>>>(W21, p1, out, nullptr, nullptr, 0);
  mlp_wmma_k<<<dim3(1579, 1, B_), 32, 0, stream>>>(W22, p2, out, nullptr, nullptr, 0);

  // --- layernorm over (C,N,T) per batch element ---
  stats_k<<<B_, 256, 0, stream>>>(out, st);
  const size_t total = (size_t)B_ * C_ * NT;
  norm_k<<<(unsigned)((total + 255) / 256), 256, 0, stream>>>(out, st, lnw, lnb, idx);
}